// SparseMoE_34772055228830
// MI455X (gfx1250) — compile-verified
//
#include <hip/hip_runtime.h>

// SparseMoE forward for MI455X (gfx1250, wave32, WMMA f32 16x16x4).
// Dense-expert evaluation (matches reference math exactly in f32),
// one kernel launch per layer, wave==expert inside each 16-token tile.

#define T_TOK 24000   // 8 * 3000 tokens
#define DIM   80
#define HID   320
#define NEXP  8
#define NLAY  4

typedef __attribute__((ext_vector_type(2))) float v2f;
typedef __attribute__((ext_vector_type(8))) float v8f;

__global__ __launch_bounds__(256)
void moe_layer_f32wmma(const float* __restrict__ in,   // [T, 80]
                       const float* __restrict__ Wg,   // [80, 8]  (layer slice)
                       const float* __restrict__ bg,   // [8]      (layer slice)
                       const float* __restrict__ W1,   // [8, 80, 320]
                       const float* __restrict__ b1,   // [8, 320]
                       const float* __restrict__ W2,   // [8, 320, 80]
                       const float* __restrict__ b2,   // [8, 80]
                       float* __restrict__ out)        // [T, 80]
{
    __shared__ float sX[16][DIM];            // input tile
    __shared__ float sW[NEXP][16];           // combine weight per (expert, token)
    __shared__ float sStage[NEXP][16][20];   // per-wave 16x16 H staging (padded)
    __shared__ float sOut[16][DIM];          // cross-expert reduction

    const int tid  = threadIdx.x;
    const int wave = tid >> 5;               // expert id, 0..7
    const int lane = tid & 31;
    const int t0   = blockIdx.x * 16;        // first token of this tile

    // ---- cooperative: load X tile, clear output accumulator ----
    for (int i = tid; i < 16 * DIM; i += 256) {
        sX[i / DIM][i % DIM]  = in[t0 * DIM + i];
        sOut[i / DIM][i % DIM] = 0.0f;
    }
    __syncthreads();

    // ---- router (redundant per wave; lanes 0..15 = one token each) ----
    if (lane < 16) {
        float lg[NEXP];
        #pragma unroll
        for (int e = 0; e < NEXP; ++e) lg[e] = bg[e];
        for (int d = 0; d < DIM; ++d) {
            const float xv = sX[lane][d];
            #pragma unroll
            for (int e = 0; e < NEXP; ++e) lg[e] += xv * Wg[d * NEXP + e];
        }
        // top-1 (lowest index wins ties, matching jax.lax.top_k)
        int i0 = 0; float v0 = lg[0];
        #pragma unroll
        for (int e = 1; e < NEXP; ++e) { if (lg[e] > v0) { v0 = lg[e]; i0 = e; } }
        // top-2
        int i1 = -1; float v1 = -3.402823466e38f;
        #pragma unroll
        for (int e = 0; e < NEXP; ++e) {
            if (e != i0 && lg[e] > v1) { v1 = lg[e]; i1 = e; }
        }
        const float w = (wave == i0) ? v0 : ((wave == i1) ? v1 : 0.0f);
        sW[wave][lane] = w;
    }
    __syncthreads();

    // ---- per-lane A fragments of X: 20 chunks of 16x4 f32 (K = 80) ----
    // ISA layout: lanes 0-15 hold M=lane, K = {4k, 4k+1}; lanes 16-31 hold K = {4k+2, 4k+3}.
    const int mrow  = lane & 15;
    const int khalf = (lane < 16) ? 0 : 2;
    const int ncol  = lane & 15;
    const int rbase = (lane < 16) ? 0 : 8;

    v2f aF[20];
    #pragma unroll
    for (int k = 0; k < 20; ++k) {
        aF[k].x = sX[mrow][4 * k + khalf];
        aF[k].y = sX[mrow][4 * k + khalf + 1];
    }

    const float* W1e = W1 + wave * DIM * HID;
    const float* W2e = W2 + wave * HID * DIM;
    const float* b1e = b1 + wave * HID;
    const float* b2e = b2 + wave * DIM;

    v8f eoAcc[5];
    #pragma unroll
    for (int nd = 0; nd < 5; ++nd)
        eoAcc[nd] = (v8f){0.f, 0.f, 0.f, 0.f, 0.f, 0.f, 0.f, 0.f};

    // ---- fused FFN: for each 16-wide hidden tile, GEMM1 -> ReLU -> GEMM2 ----
    for (int nh = 0; nh < 20; ++nh) {
        // GEMM1: Htile(16x16) = X(16x80) * W1[:, nh*16 : nh*16+16]
        v8f hAcc = (v8f){0.f, 0.f, 0.f, 0.f, 0.f, 0.f, 0.f, 0.f};
        #pragma unroll
        for (int k = 0; k < 20; ++k) {
            const int kk = 4 * k + khalf;
            v2f b;
            b.x = W1e[(kk    ) * HID + nh * 16 + ncol];
            b.y = W1e[(kk + 1) * HID + nh * 16 + ncol];
            hAcc = __builtin_amdgcn_wmma_f32_16x16x4_f32(
                false, aF[k], false, b, (short)0, hAcc, false, false);
        }

        // bias + ReLU in C/D layout (element r -> token rbase + r)
        const float bias1 = b1e[nh * 16 + ncol];
        #pragma unroll
        for (int r = 0; r < 8; ++r) {
            const float v = hAcc[r] + bias1;
            hAcc[r] = v > 0.0f ? v : 0.0f;
        }

        // C/D-layout -> A-layout transpose bounce via wave-private LDS
        #pragma unroll
        for (int r = 0; r < 8; ++r)
            sStage[wave][rbase + r][ncol] = hAcc[r];
        asm volatile("s_wait_dscnt 0" ::: "memory");

        v2f hF[4];
        #pragma unroll
        for (int kk2 = 0; kk2 < 4; ++kk2) {
            hF[kk2].x = sStage[wave][mrow][4 * kk2 + khalf];
            hF[kk2].y = sStage[wave][mrow][4 * kk2 + khalf + 1];
        }
        asm volatile("s_wait_dscnt 0" ::: "memory");

        // GEMM2: EO(16x80) += Htile(16x16) * W2[nh*16 : nh*16+16, :]
        #pragma unroll
        for (int nd = 0; nd < 5; ++nd) {
            #pragma unroll
            for (int kk2 = 0; kk2 < 4; ++kk2) {
                const int kg = nh * 16 + 4 * kk2 + khalf;
                v2f b;
                b.x = W2e[(kg    ) * DIM + nd * 16 + ncol];
                b.y = W2e[(kg + 1) * DIM + nd * 16 + ncol];
                eoAcc[nd] = __builtin_amdgcn_wmma_f32_16x16x4_f32(
                    false, hF[kk2], false, b, (short)0, eoAcc[nd], false, false);
            }
        }
    }

    // ---- combine: (EO + b2) * w[token] ; reduce over experts via ds_add_f32 ----
    #pragma unroll
    for (int nd = 0; nd < 5; ++nd) {
        const float bias2 = b2e[nd * 16 + ncol];
        #pragma unroll
        for (int r = 0; r < 8; ++r) {
            const int row = rbase + r;
            const float val = (eoAcc[nd][r] + bias2) * sW[wave][row];
            atomicAdd(&sOut[row][nd * 16 + ncol], val);
        }
    }
    __syncthreads();

    for (int i = tid; i < 16 * DIM; i += 256)
        out[t0 * DIM + i] = sOut[i / DIM][i % DIM];
}

extern "C" void kernel_launch(void* const* d_in, const int* in_sizes, int n_in,
                              void* d_out, int out_size, void* d_ws, size_t ws_size,
                              hipStream_t stream) {
    (void)in_sizes; (void)n_in; (void)out_size; (void)ws_size;
    const float* x  = (const float*)d_in[0];
    const float* Wg = (const float*)d_in[1];   // [4, 80, 8]
    const float* bg = (const float*)d_in[2];   // [4, 8]
    const float* W1 = (const float*)d_in[3];   // [8, 80, 320]
    const float* b1 = (const float*)d_in[4];   // [8, 320]
    const float* W2 = (const float*)d_in[5];   // [8, 320, 80]
    const float* b2 = (const float*)d_in[6];   // [8, 80]
    float* outp = (float*)d_out;

    float* buf0 = (float*)d_ws;                // ping-pong activation buffers
    float* buf1 = buf0 + (size_t)T_TOK * DIM;  // needs 2 * 7.68 MB of ws

    const int nTiles = T_TOK / 16;             // 1500 workgroups
    const float* src = x;
    for (int l = 0; l < NLAY; ++l) {
        float* dst = (l == NLAY - 1) ? outp : ((l & 1) ? buf1 : buf0);
        moe_layer_f32wmma<<<nTiles, 256, 0, stream>>>(
            src, Wg + (size_t)l * DIM * NEXP, bg + (size_t)l * NEXP,
            W1, b1, W2, b2, dst);
        src = dst;
    }
}